// MultiBoxFocalLoss_75763223101576
// MI455X (gfx1250) — compile-verified
//
#include <hip/hip_runtime.h>
#include <stdint.h>

#define NUM_B 32
#define NUM_P 24564
#define NUM_T 20
#define NUM_C 81
#define ROWS_TOTAL (NUM_B * NUM_P)                     // 786048
#define TILE_ROWS 64
#define NTILES ((ROWS_TOTAL + TILE_ROWS - 1) / TILE_ROWS)  // 12282 (exact)

typedef __attribute__((ext_vector_type(4))) unsigned int uint32x4;
typedef __attribute__((ext_vector_type(8))) int          int32x8;
typedef __attribute__((ext_vector_type(4))) int          int32x4;
typedef __attribute__((ext_vector_type(2))) float        v2f;
typedef __attribute__((ext_vector_type(8))) float        v8f;

// ---------------------------------------------------------------- workspace
// [0      .. 5120)  : u64 bp_best[B*T]   (packed {iou_bits, prior_idx})
// [5120   .. 5128)  : float loss_l, loss_c
// [5128   .. 5132)  : u32 n_pos
// [8192   ..  +3.1M): int conf_t[ROWS_TOTAL]

__device__ __forceinline__ float smooth_l1(float d) {
    float a = fabsf(d);
    return (a < 1.0f) ? 0.5f * d * d : a - 0.5f;
}

// ---------------------------------------------------------------- K0: init
__global__ void k_init(unsigned long long* bp, float* acc, unsigned* npos) {
    int i = blockIdx.x * blockDim.x + threadIdx.x;
    if (i < NUM_B * NUM_T) bp[i] = 0ull;
    if (i == 0) { acc[0] = 0.0f; acc[1] = 0.0f; npos[0] = 0u; }
}

// ---------------------------------------------------------------- K1: per-truth best prior (argmax over P)
__global__ void __launch_bounds__(256) k_match(const float* __restrict__ priors,
                                               const float* __restrict__ targets,
                                               unsigned long long* __restrict__ bp_best) {
    __shared__ float tx0[NUM_T], ty0[NUM_T], tx1[NUM_T], ty1[NUM_T], tarea[NUM_T];
    __shared__ unsigned long long lred[NUM_T];
    const int tid = threadIdx.x;
    const int b = blockIdx.y;
    if (tid < NUM_T) {
        const float* tp = targets + (size_t)(b * NUM_T + tid) * 5;
        float x0 = tp[0], y0 = tp[1], x1 = tp[2], y1 = tp[3];
        tx0[tid] = x0; ty0[tid] = y0; tx1[tid] = x1; ty1[tid] = y1;
        tarea[tid] = (x1 - x0) * (y1 - y0);
        lred[tid] = 0ull;
    }
    __syncthreads();
    int p = blockIdx.x * blockDim.x + tid;
    if (p < NUM_P) {
        float pcx = priors[p * 4 + 0], pcy = priors[p * 4 + 1];
        float pw  = priors[p * 4 + 2], ph  = priors[p * 4 + 3];
        float px0 = pcx - 0.5f * pw, py0 = pcy - 0.5f * ph;
        float px1 = pcx + 0.5f * pw, py1 = pcy + 0.5f * ph;
        float parea = pw * ph;
        #pragma unroll 4
        for (int t = 0; t < NUM_T; ++t) {
            float lx = fmaxf(px0, tx0[t]), ly = fmaxf(py0, ty0[t]);
            float rx = fminf(px1, tx1[t]), ry = fminf(py1, ty1[t]);
            float w = fmaxf(rx - lx, 0.0f), h = fmaxf(ry - ly, 0.0f);
            float inter = w * h;
            float iou = inter / (tarea[t] + parea - inter);
            unsigned long long key =
                ((unsigned long long)__float_as_uint(iou) << 32) | (unsigned)p;
            atomicMax(&lred[t], key);
        }
    }
    __syncthreads();
    if (tid < NUM_T) atomicMax(&bp_best[b * NUM_T + tid], lred[tid]);
}

// ---------------------------------------------------------------- K2: per-prior match + encode + smooth-L1
__global__ void __launch_bounds__(256) k_encode_loc(const float* __restrict__ loc_data,
                                                    const float* __restrict__ priors,
                                                    const float* __restrict__ targets,
                                                    const unsigned long long* __restrict__ bp_best,
                                                    int* __restrict__ conf_t,
                                                    float* __restrict__ acc,
                                                    unsigned* __restrict__ npos) {
    __shared__ float tx0[NUM_T], ty0[NUM_T], tx1[NUM_T], ty1[NUM_T], tarea[NUM_T], tlab[NUM_T];
    __shared__ unsigned bpj[NUM_T];
    __shared__ float redf[256];
    __shared__ unsigned redi[256];
    const int tid = threadIdx.x;
    const int b = blockIdx.y;
    if (tid < NUM_T) {
        const float* tp = targets + (size_t)(b * NUM_T + tid) * 5;
        float x0 = tp[0], y0 = tp[1], x1 = tp[2], y1 = tp[3];
        tx0[tid] = x0; ty0[tid] = y0; tx1[tid] = x1; ty1[tid] = y1;
        tarea[tid] = (x1 - x0) * (y1 - y0);
        tlab[tid] = tp[4];
        bpj[tid] = (unsigned)(bp_best[b * NUM_T + tid] & 0xffffffffull);
    }
    __syncthreads();
    int p = blockIdx.x * blockDim.x + tid;
    float ll = 0.0f;
    unsigned np = 0u;
    if (p < NUM_P) {
        float pcx = priors[p * 4 + 0], pcy = priors[p * 4 + 1];
        float pw  = priors[p * 4 + 2], ph  = priors[p * 4 + 3];
        float px0 = pcx - 0.5f * pw, py0 = pcy - 0.5f * ph;
        float px1 = pcx + 0.5f * pw, py1 = pcy + 0.5f * ph;
        float parea = pw * ph;
        float bto = -1.0f; int bti = 0;
        #pragma unroll 4
        for (int t = 0; t < NUM_T; ++t) {
            float lx = fmaxf(px0, tx0[t]), ly = fmaxf(py0, ty0[t]);
            float rx = fminf(px1, tx1[t]), ry = fminf(py1, ty1[t]);
            float w = fmaxf(rx - lx, 0.0f), h = fmaxf(ry - ly, 0.0f);
            float inter = w * h;
            float iou = inter / (tarea[t] + parea - inter);
            if (iou > bto) { bto = iou; bti = t; }
        }
        // best-prior override: best_truth_overlap[bp_idx[t]] = 2.0, idx = t
        #pragma unroll 4
        for (int t = 0; t < NUM_T; ++t)
            if (bpj[t] == (unsigned)p) { bto = 2.0f; bti = t; }
        int c = (int)tlab[bti] + 1;
        if (bto < 0.5f) c = -1;
        if (bto < 0.4f) c = 0;
        conf_t[(size_t)b * NUM_P + p] = c;
        if (c > 0) {
            np = 1u;
            float mx0 = tx0[bti], my0 = ty0[bti], mx1 = tx1[bti], my1 = ty1[bti];
            float gcx = ((mx0 + mx1) * 0.5f - pcx) / (0.1f * pw);
            float gcy = ((my0 + my1) * 0.5f - pcy) / (0.1f * ph);
            float gw = __logf((mx1 - mx0) / pw) * 5.0f;   // / VARIANCE[1]=0.2
            float gh = __logf((my1 - my0) / ph) * 5.0f;
            const float* ld = loc_data + ((size_t)b * NUM_P + p) * 4;
            ll = smooth_l1(ld[0] - gcx) + smooth_l1(ld[1] - gcy) +
                 smooth_l1(ld[2] - gw)  + smooth_l1(ld[3] - gh);
        }
    }
    redf[tid] = ll; redi[tid] = np;
    __syncthreads();
    for (int s = 128; s > 0; s >>= 1) {
        if (tid < s) { redf[tid] += redf[tid + s]; redi[tid] += redi[tid + s]; }
        __syncthreads();
    }
    if (tid == 0) { atomicAdd(&acc[0], redf[0]); atomicAdd(npos, redi[0]); }
}

// ---------------------------------------------------------------- TDM descriptor issue (2D tile: 64 rows x 81 f32)
__device__ __forceinline__ void tdm_issue_tile(const float* gbase, int tile, const float* lds_dst) {
    unsigned long long gaddr = (unsigned long long)(uintptr_t)gbase +
                               (unsigned long long)tile * (TILE_ROWS * NUM_C * 4ull);
    unsigned lds = (unsigned)(uintptr_t)lds_dst;          // low 32b of flat = LDS byte offset
    unsigned rows_rem = (unsigned)(ROWS_TOTAL - tile * TILE_ROWS);
    uint32x4 g0;
    g0.x = 1u;                                            // count=1, gather off
    g0.y = lds;                                           // lds_addr
    g0.z = (unsigned)(gaddr & 0xffffffffull);             // global_addr[31:0]
    g0.w = (unsigned)((gaddr >> 32) & 0x1ffffffull) | (2u << 30);   // addr[56:32] | type=2
    int32x8 g1;
    g1.s0 = (int)(2u << 16);                              // data_size = 4B
    g1.s1 = (int)((NUM_C & 0xffffu) << 16);               // tensor_dim0[15:0]
    g1.s2 = (int)((NUM_C >> 16) | ((rows_rem & 0xffffu) << 16));   // dim0 hi | dim1 lo
    g1.s3 = (int)((rows_rem >> 16) | ((unsigned)NUM_C << 16));     // dim1 hi | tile_dim0=81
    g1.s4 = (int)(TILE_ROWS);                             // tile_dim1=64, tile_dim2=0
    g1.s5 = (int)NUM_C;                                   // tensor_dim0_stride = 81
    g1.s6 = 0; g1.s7 = 0;
    int32x4 g2 = {0, 0, 0, 0};                            // tensor_dim2.. = 0 (2D)
    int32x4 g3 = {0, 0, 0, 0};
    int32x8 g4 = {0, 0, 0, 0, 0, 0, 0, 0};                // 6-arg toolchain: extra group, unused
    __builtin_amdgcn_tensor_load_to_lds(g0, g1, g2, g3, g4, 0);
}

// ---------------------------------------------------------------- K3: focal loss (TDM double-buffer + WMMA sum-exp)
// 128 threads = 4 waves; wave w owns 16 rows of the 64-row tile.
__global__ void __launch_bounds__(128) k_focal(const float* __restrict__ conf_data,
                                               const int* __restrict__ conf_t,
                                               float* __restrict__ acc) {
    __shared__ float smem[2][TILE_ROWS * NUM_C];   // 2 x 20736 B
    __shared__ float xchg[4][16];
    __shared__ float red[128];
    const int tid = threadIdx.x;
    const int wid = tid >> 5;
    const int lane = tid & 31;
    float acc_c = 0.0f;

    int tile = (int)blockIdx.x;
    if (tile < NTILES && wid == 0) tdm_issue_tile(conf_data, tile, &smem[0][0]);
    int buf = 0;
    for (; tile < NTILES; tile += (int)gridDim.x) {
        int nxt = tile + (int)gridDim.x;
        if (wid == 0) {
            if (nxt < NTILES) {
                tdm_issue_tile(conf_data, nxt, &smem[buf ^ 1][0]);
                __builtin_amdgcn_s_wait_tensorcnt(1);    // current tile done (TDM in-order)
            } else {
                __builtin_amdgcn_s_wait_tensorcnt(0);
            }
        }
        __syncthreads();

        const float* rows = &smem[buf][0];
        const int rbase = wid * 16;
        const int rhalf = lane & 15;                 // row within group for this lane pair
        const float* rp = rows + (rbase + rhalf) * NUM_C;

        // row max: lanes l and l+16 split the 81 classes, then swap halves
        float m = -3.0e38f;
        for (int c = (lane >> 4); c < NUM_C; c += 2) m = fmaxf(m, rp[c]);
        m = fmaxf(m, __shfl_xor(m, 16));

        // sum(exp(x - max)) per row via WMMA: A = 16x4 exp tile, B = ones(4x16)
        v2f ones; ones.x = 1.0f; ones.y = 1.0f;
        v8f sacc = {};
        const int cb = (lane >> 4) * 2;              // K-pair offset for this half-wave
        #pragma unroll
        for (int k = 0; k < 21; ++k) {               // 21*4 = 84 >= 81, tail padded with 0
            int c0 = 4 * k + cb;
            int c1 = c0 + 1;
            v2f a;
            a.x = (c0 < NUM_C) ? __expf(rp[c0] - m) : 0.0f;
            a.y = (c1 < NUM_C) ? __expf(rp[c1] - m) : 0.0f;
            sacc = __builtin_amdgcn_wmma_f32_16x16x4_f32(
                false, a, false, ones, (short)0, sacc, false, false);
        }
        // D layout: lanes 0-15 hold rows 0..7 in comps 0..7; lanes 16-31 rows 8..15
        if (lane == 0) {
            #pragma unroll
            for (int i = 0; i < 8; ++i) xchg[wid][i] = sacc[i];
        }
        if (lane == 16) {
            #pragma unroll
            for (int i = 0; i < 8; ++i) xchg[wid][8 + i] = sacc[i];
        }
        asm volatile("s_wait_dscnt 0" ::: "memory"); // DS in-order within wave; block reordering

        if (lane < 16) {
            float s = xchg[wid][lane];
            int grow = tile * TILE_ROWS + rbase + lane;
            if (grow < ROWS_TOTAL) {
                int tgt = conf_t[grow];
                if (tgt >= 0) {
                    float xt = rows[(rbase + lane) * NUM_C + tgt];
                    float logpt = (xt - m) - __logf(s);
                    float pt = __expf(logpt);
                    float at = (tgt > 0) ? 0.25f : 0.75f;
                    float omp = 1.0f - pt;
                    acc_c += -at * omp * omp * logpt;
                }
            }
        }
        __syncthreads();
        buf ^= 1;
    }

    red[tid] = acc_c;
    __syncthreads();
    for (int s = 64; s > 0; s >>= 1) {
        if (tid < s) red[tid] += red[tid + s];
        __syncthreads();
    }
    if (tid == 0) atomicAdd(&acc[1], red[0]);
}

// ---------------------------------------------------------------- K4: finalize
__global__ void k_final(const float* __restrict__ acc, const unsigned* __restrict__ npos,
                        float* __restrict__ out) {
    float n = (float)npos[0];
    out[0] = acc[0] / n;
    out[1] = acc[1] / n;
}

// ---------------------------------------------------------------- launch
extern "C" void kernel_launch(void* const* d_in, const int* in_sizes, int n_in,
                              void* d_out, int out_size, void* d_ws, size_t ws_size,
                              hipStream_t stream) {
    const float* loc    = (const float*)d_in[0];
    const float* conf   = (const float*)d_in[1];
    const float* priors = (const float*)d_in[2];
    const float* tgts   = (const float*)d_in[3];

    char* ws = (char*)d_ws;
    unsigned long long* bp = (unsigned long long*)ws;           // 640 * 8 B
    float*    acc  = (float*)(ws + 5120);                        // loss_l, loss_c
    unsigned* npos = (unsigned*)(ws + 5128);
    int*      conf_t = (int*)(ws + 8192);                        // ROWS_TOTAL ints

    k_init<<<dim3(3), dim3(256), 0, stream>>>(bp, acc, npos);

    dim3 gmatch((NUM_P + 255) / 256, NUM_B);
    k_match<<<gmatch, dim3(256), 0, stream>>>(priors, tgts, bp);
    k_encode_loc<<<gmatch, dim3(256), 0, stream>>>(loc, priors, tgts, bp, conf_t, acc, npos);

    int nblk = 2048;
    if (nblk > NTILES) nblk = NTILES;
    k_focal<<<dim3(nblk), dim3(128), 0, stream>>>(conf, conf_t, acc);

    k_final<<<1, 1, 0, stream>>>(acc, npos, (float*)d_out);
}